// RLMahjong_9113920602297
// MI455X (gfx1250) — compile-verified
//
#include <hip/hip_runtime.h>
#include <hip/hip_bf16.h>
#include <cstdint>

// ---------------------------------------------------------------------------
// Retrace(lambda) targets + advantages for MI455X (gfx1250).
//
// Roofline: ~430 MB of unavoidable HBM traffic @ 23.3 TB/s => ~19 us; compute
// is ~0.12 GFLOP => pure streaming problem.  Design:
//   K1: stage 16x54 q/pi tiles into LDS with GLOBAL_LOAD_ASYNC_TO_LDS_B64
//       (ASYNCcnt path, no VGPR round-trip), compute v_t = pi.q via
//       V_WMMA_F32_16X16X4_F32 (ones-matrix trick), extract q[t,a_t],
//       pi[t,a_t], and emit advantages with NT stores.
//   K2: per-t scalars (qe, delta, coef) + per-block (2048 t) affine summary
//       of the backward recurrence y <- a*y + b  (a = gl*clip(rho)*(1-done),
//       b = delta), composed via per-thread serial + LDS tree reduction.
//   K3: sequential scan over the 256 block summaries -> per-block carry.
//   K4: per-block replay: thread-serial compose (8 t) + Hillis-Steele affine
//       scan over 256 threads + replay to produce g and targets_q.
// ---------------------------------------------------------------------------

#define A_DIM 54
#define APAD  56
#define GAMMA 0.999f
#define LAMBD 0.95f
#define ALPHA 0.99f
#define WAVES 8
#define CHUNK 2048
#define SUB   8

typedef float v2f __attribute__((ext_vector_type(2)));
typedef float v8f __attribute__((ext_vector_type(8)));

// ---------------- Kernel 1: v, q_t_a, pi_t_a, advantages -------------------
__global__ __launch_bounds__(256) void k1_v_adv(
    const float* __restrict__ q, const float* __restrict__ pi,
    const int* __restrict__ a_t,
    float* __restrict__ vws, float* __restrict__ qta, float* __restrict__ pita,
    float* __restrict__ adv, int T, int numTiles)
{
    __shared__ __align__(16) float sQ[WAVES][16 * APAD];
    __shared__ __align__(16) float sP[WAVES][16 * APAD];
    __shared__ float sV[WAVES][16];

    const int lane = threadIdx.x & 31;
    const int wid  = threadIdx.x >> 5;
    const int tile = blockIdx.x * WAVES + wid;
    // Invalid tiles still run (uniform barriers) but all rows are OOB.
    const long t0 = (tile < numTiles) ? (long)tile * 16 : (long)T + 1;

    // Zero the two pad columns (54,55) of every row (read by WMMA chunk 13).
    {
        v2f z = {0.f, 0.f};
        int r = lane & 15;
        if (lane < 16) *(v2f*)&sQ[wid][r * APAD + A_DIM] = z;
        else           *(v2f*)&sP[wid][r * APAD + A_DIM] = z;
    }

    // Stage 16x54 tiles straight into LDS via the CDNA5 async-copy path.
    // GVS addressing: saddr = array base, vaddr = u32 byte offset (<2^31).
    const uint32_t ldsQ = (uint32_t)(uintptr_t)&sQ[wid][0];
    const uint32_t ldsP = (uint32_t)(uintptr_t)&sP[wid][0];
    const uint32_t gq   = (uint32_t)((uint64_t)t0 * (A_DIM * 4));
    #pragma unroll
    for (int it = 0; it < 14; ++it) {
        int e = it * 64 + lane * 2;              // even => 8B aligned pairs
        if (e < 16 * A_DIM) {
            int r = e / A_DIM;
            int c = e - r * A_DIM;               // even
            long t = t0 + r;
            if (t <= (long)T) {
                uint32_t goff = gq + (uint32_t)e * 4u;
                uint32_t lq = ldsQ + (uint32_t)(r * APAD + c) * 4u;
                uint32_t lp = ldsP + (uint32_t)(r * APAD + c) * 4u;
                asm volatile("global_load_async_to_lds_b64 %0, %1, %2"
                             :: "v"(lq), "v"(goff),
                                "s"((uint64_t)(uintptr_t)q) : "memory");
                asm volatile("global_load_async_to_lds_b64 %0, %1, %2"
                             :: "v"(lp), "v"(goff),
                                "s"((uint64_t)(uintptr_t)pi) : "memory");
            }
        }
    }
    asm volatile("s_wait_asynccnt 0" ::: "memory");
    __syncthreads();

    // v[0..15] = rowsum(pi .* q) via D = (pi.*q)[16x56] x Ones[56x16].
    // A-operand layout for V_WMMA_F32_16X16X4_F32: lane<16 holds K={0,1},
    // lane>=16 holds K={2,3} of row (lane&15).
    const int rowA = lane & 15;
    const int kOff = (lane >> 4) * 2;
    const float* bq = &sQ[wid][rowA * APAD + kOff];
    const float* bp = &sP[wid][rowA * APAD + kOff];
    v8f acc = {0.f, 0.f, 0.f, 0.f, 0.f, 0.f, 0.f, 0.f};
    const v2f ones = {1.f, 1.f};
    #pragma unroll
    for (int c4 = 0; c4 < 14; ++c4) {
        v2f qv = *(const v2f*)(bq + c4 * 4);
        v2f pv = *(const v2f*)(bp + c4 * 4);
        v2f a  = qv * pv;
        acc = __builtin_amdgcn_wmma_f32_16x16x4_f32(
            /*neg_a=*/false, a, /*neg_b=*/false, ones,
            /*c_mod=*/(short)0, acc, /*reuse_a=*/false, /*reuse_b=*/false);
    }
    // Every lane<16 holds v[0..7] in acc, lanes>=16 hold v[8..15].
    if ((lane & 15) == 0) {
        int vb = (lane >> 4) * 8;
        #pragma unroll
        for (int g2 = 0; g2 < 8; ++g2) sV[wid][vb + g2] = acc[g2];
    }
    __syncthreads();

    // v, q[t,a], pi[t,a] extraction (avoids later gathers over q/pi).
    if (lane < 16) {
        long t = t0 + lane;
        if (t <= (long)T) vws[t] = sV[wid][lane];
        if (t <  (long)T) {
            int a  = a_t[t];
            qta[t]  = sQ[wid][lane * APAD + a];
            pita[t] = sP[wid][lane * APAD + a];
        }
    }

    // advantages = (1-alpha) * (q_t - v_t), streamed with NT stores.
    #pragma unroll
    for (int it = 0; it < 27; ++it) {
        int e = it * 32 + lane;              // 0..863
        int r = e / A_DIM;
        int c = e - r * A_DIM;
        long t = t0 + r;
        if (t < (long)T) {
            float val = (1.0f - ALPHA) * (sQ[wid][r * APAD + c] - sV[wid][r]);
            __builtin_nontemporal_store(val, adv + t * A_DIM + c);
        }
    }
}

// ------------- Kernel 2: per-t scalars + per-block affine summary ----------
__global__ __launch_bounds__(256) void k2_scalars(
    const float* __restrict__ r_t, const float* __restrict__ done_t,
    const int* __restrict__ a_t, const float* __restrict__ mu,
    const float* __restrict__ vws, const float* __restrict__ qta,
    const float* __restrict__ pita,
    float* __restrict__ qe, float* __restrict__ delta, float* __restrict__ coef,
    float* __restrict__ blkA, float* __restrict__ blkB, int T)
{
    __shared__ float sA[256], sB[256];
    const int j = threadIdx.x;
    const long base = (long)blockIdx.x * CHUNK + (long)j * SUB;

    float Acc = 1.f, Bcc = 0.f;              // identity affine
    #pragma unroll
    for (int k = SUB - 1; k >= 0; --k) {     // descending t = scan order
        long t = base + k;
        if (t < (long)T) {
            float d   = done_t[t];
            float qev = r_t[t] + (1.f - d) * GAMMA * vws[t + 1];
            int   a   = a_t[t];
            float rho = pita[t] / mu[t * A_DIM + a];
            float td  = qev - qta[t];
            float del = LAMBD * GAMMA * rho * td;
            float cf  = GAMMA * LAMBD * fminf(fmaxf(rho, 0.f), 1.f);
            qe[t] = qev; delta[t] = del; coef[t] = cf;
            float aa = (d > 0.5f) ? 0.f : cf;
            Acc = aa * Acc;
            Bcc = aa * Bcc + del;
        }
    }
    sA[j] = Acc; sB[j] = Bcc;
    // Tree-compose in descending-thread order: M = M_j o M_{j+s}.
    for (int s = 128; s >= 1; s >>= 1) {
        __syncthreads();
        if (j < s) {
            float A1 = sA[j], B1 = sB[j], A2 = sA[j + s], B2 = sB[j + s];
            sA[j] = A1 * A2;
            sB[j] = A1 * B2 + B1;
        }
    }
    if (j == 0) { blkA[blockIdx.x] = sA[0]; blkB[blockIdx.x] = sB[0]; }
}

// ------------- Kernel 3: sequential scan over block summaries --------------
__global__ __launch_bounds__(256) void k3_blockscan(
    const float* __restrict__ blkA, const float* __restrict__ blkB,
    float* __restrict__ carry, int nb)
{
    __shared__ float sA[256], sB[256], sC[256];
    const int j = threadIdx.x;
    if (j < nb) { sA[j] = blkA[j]; sB[j] = blkB[j]; }
    __syncthreads();
    if (j == 0) {
        float y = 0.f;
        for (int b = nb - 1; b >= 0; --b) {
            sC[b] = y;                       // carry entering block b
            y = sA[b] * y + sB[b];
        }
    }
    __syncthreads();
    if (j < nb) carry[j] = sC[j];
}

// ------------- Kernel 4: replay scan, emit targets_q -----------------------
__global__ __launch_bounds__(256) void k4_apply(
    const float* __restrict__ done_t, const float* __restrict__ qe,
    const float* __restrict__ delta, const float* __restrict__ coef,
    const float* __restrict__ qta, const float* __restrict__ vws,
    const float* __restrict__ carry, float* __restrict__ targets, int T)
{
    __shared__ float sA[256], sB[256];
    const int j = threadIdx.x;
    const long base = (long)blockIdx.x * CHUNK + (long)j * SUB;

    float dk[SUB], cf[SUB], dl[SUB];
    float Acc = 1.f, Bcc = 0.f;
    #pragma unroll
    for (int k = SUB - 1; k >= 0; --k) {
        long t = base + k;
        bool valid = t < (long)T;
        float d  = valid ? done_t[t] : 0.f;
        float c  = valid ? coef[t]   : 1.f;   // identity for OOB
        float de = valid ? delta[t]  : 0.f;
        dk[k] = d; cf[k] = c; dl[k] = de;
        float aa = (d > 0.5f) ? 0.f : c;
        Acc = aa * Acc;
        Bcc = aa * Bcc + de;
    }
    sA[j] = Acc; sB[j] = Bcc;
    // Hillis-Steele inclusive scan: S_j = M_j o M_{j+1} o ... o M_255.
    for (int s = 1; s < 256; s <<= 1) {
        __syncthreads();
        float A1 = sA[j], B1 = sB[j];
        float A2 = 1.f, B2 = 0.f;
        if (j + s < 256) { A2 = sA[j + s]; B2 = sB[j + s]; }
        __syncthreads();
        sA[j] = A1 * A2;
        sB[j] = A1 * B2 + B1;
    }
    __syncthreads();

    const float y0 = carry[blockIdx.x];
    // Exclusive carry into thread j: S_{j+1}(y0).
    float Aex = (j < 255) ? sA[j + 1] : 1.f;
    float Bex = (j < 255) ? sB[j + 1] : 0.f;
    float y = Aex * y0 + Bex;

    #pragma unroll
    for (int k = SUB - 1; k >= 0; --k) {
        long t = base + k;
        if (t < (long)T) {
            float yu = (dk[k] > 0.5f) ? 0.f : y;
            float g  = qe[t] + yu;
            targets[t] = g + ALPHA * (qta[t] - vws[t]);
            y = dl[k] + cf[k] * yu;
        }
    }
}

// ---------------------------------------------------------------------------
extern "C" void kernel_launch(void* const* d_in, const int* in_sizes, int n_in,
                              void* d_out, int out_size, void* d_ws, size_t ws_size,
                              hipStream_t stream)
{
    (void)n_in; (void)out_size; (void)ws_size;
    const float* q    = (const float*)d_in[0];   // (T+1, 54)
    const float* pi   = (const float*)d_in[1];   // (T+1, 54)
    const int*   a_t  = (const int*)  d_in[2];   // (T,)
    const float* r_t  = (const float*)d_in[3];   // (T,)
    const float* mu   = (const float*)d_in[4];   // (T, 54)
    const float* done = (const float*)d_in[5];   // (T,)
    const int T = in_sizes[3];

    float* targets = (float*)d_out;              // (T,)
    float* adv     = targets + T;                // (T, 54)

    float* wsf   = (float*)d_ws;
    float* vws   = wsf;                          // T+1
    float* qta   = vws  + (T + 1);               // T
    float* pita  = qta  + T;                     // T
    float* qe    = pita + T;                     // T
    float* delta = qe   + T;                     // T
    float* coef  = delta + T;                    // T
    const int nb = (T + CHUNK - 1) / CHUNK;      // 256 for T = 524288
    float* blkA  = coef + T;
    float* blkB  = blkA + nb;
    float* carry = blkB + nb;

    const int numTiles = (T + 1 + 15) / 16;
    const int grid1 = (numTiles + WAVES - 1) / WAVES;

    k1_v_adv   <<<grid1, 256, 0, stream>>>(q, pi, a_t, vws, qta, pita, adv, T, numTiles);
    k2_scalars <<<nb,    256, 0, stream>>>(r_t, done, a_t, mu, vws, qta, pita,
                                           qe, delta, coef, blkA, blkB, T);
    k3_blockscan<<<1,    256, 0, stream>>>(blkA, blkB, carry, nb);
    k4_apply   <<<nb,    256, 0, stream>>>(done, qe, delta, coef, qta, vws,
                                           carry, targets, T);
}